// DynamycMoE_49959059587400
// MI455X (gfx1250) — compile-verified
//
#include <hip/hip_runtime.h>
#include <math.h>
#include <stdint.h>

// Problem constants (match reference)
#define B_TOK 8192
#define DIM   768
#define HID   256
#define NE    8
#define NC    64
#define NT    512
#define TM    32     // tokens per block tile in the main kernel
#define NCHUNK (DIM / 32)

typedef float v2f __attribute__((ext_vector_type(2)));
typedef float v8f __attribute__((ext_vector_type(8)));
typedef unsigned int v4u __attribute__((ext_vector_type(4)));
typedef unsigned int v8u __attribute__((ext_vector_type(8)));

// Must match the builtin's parameter type exactly (gcc-style vector).
typedef int v4i_vs __attribute__((__vector_size__(16)));
typedef __attribute__((address_space(1))) v4i_vs glb_v4i;
typedef __attribute__((address_space(3))) v4i_vs lds_v4i;

#define WMMA_F32(a, b, c) \
  __builtin_amdgcn_wmma_f32_16x16x4_f32(false, (a), false, (b), (short)0, (c), false, false)

// ---- CDNA5 async global->LDS copy (ASYNCcnt) with safe fallback ------------
#if __has_builtin(__builtin_amdgcn_global_load_async_to_lds_b128)
#define HAVE_ASYNC 1
#else
#define HAVE_ASYNC 0
#endif

static __device__ __forceinline__ void async_copy16(const float* g, float* l) {
#if HAVE_ASYNC
  __builtin_amdgcn_global_load_async_to_lds_b128(
      (glb_v4i*)(uintptr_t)g,
      (lds_v4i*)(unsigned)(uintptr_t)l, 0, 0);
#else
  *(float4*)l = *(const float4*)g;
#endif
}

#if HAVE_ASYNC
#if __has_builtin(__builtin_amdgcn_s_wait_asynccnt)
#define WAIT_ASYNC(n) __builtin_amdgcn_s_wait_asynccnt(n)
#else
#define WAIT_ASYNC(n) asm volatile("s_wait_asynccnt " #n ::: "memory")
#endif
#else
#define WAIT_ASYNC(n)
#endif

#if __has_builtin(__builtin_amdgcn_s_wait_tensorcnt)
#define WAIT_TENSOR(n) __builtin_amdgcn_s_wait_tensorcnt(n)
#else
#define WAIT_TENSOR(n) asm volatile("s_wait_tensorcnt " #n ::: "memory")
#endif

// ---- Tensor Data Mover: 2-D tile load, global -> LDS (TENSORcnt) -----------
// D# group0 (4 SGPRs) + group1 (8 SGPRs); groups 2/3 omitted (2-D tensor).
// LDS row padding: pad_interval v -> 2^(v+1) DWORDs of data, then
// (pad_amount+1) DWORDs of pad.
static __device__ __forceinline__ void tdm_load_2d(const float* gbase,
                                                   unsigned lds_off,
                                                   unsigned d0, unsigned d1,
                                                   unsigned stride0,
                                                   unsigned pad_interval,
                                                   unsigned pad_amount) {
  const unsigned long long ga = (unsigned long long)(uintptr_t)gbase;
  v4u g0;
  g0.x = 1u;                                               // count=1 (valid D#)
  g0.y = lds_off;                                          // lds_addr (bytes)
  g0.z = (unsigned)ga;                                     // global_addr[31:0]
  g0.w = (unsigned)((ga >> 32) & 0x01FFFFFFu) | (2u << 30); // ga[56:32] | type=2
  v8u g1;
  g1.s0 = (2u << 16) | (1u << 20) | (pad_interval << 22) | (pad_amount << 25);
  g1.s1 = (d0 & 0xFFFFu) << 16;                            // tensor_dim0 lo
  g1.s2 = (d0 >> 16) | ((d1 & 0xFFFFu) << 16);             // tensor_dim0 hi | tensor_dim1 lo
  g1.s3 = (d1 >> 16) | (d0 << 16);                         // tensor_dim1 hi | tile_dim0
  g1.s4 = d1;                                              // tile_dim1 (tile_dim2 = 0)
  g1.s5 = stride0;                                         // tensor_dim0_stride lo
  g1.s6 = 0u;
  g1.s7 = 0u;
  asm volatile("tensor_load_to_lds %0, %1" :: "s"(g0), "s"(g1) : "memory");
}

// ---------------------------------------------------------------------------
// Kernel 0: zero output accumulator and per-expert counters
// ---------------------------------------------------------------------------
__global__ __launch_bounds__(256) void zero_kernel(float* __restrict__ y,
                                                   int* __restrict__ counts, int n) {
  int i = blockIdx.x * 256 + threadIdx.x;
  if (i < n) y[i] = 0.0f;
  if (i < NE) counts[i] = 0;
}

// ---------------------------------------------------------------------------
// Kernel 1: gating (wave32 per token): logits, top-2, softmax, list append
// ---------------------------------------------------------------------------
__global__ __launch_bounds__(256) void gate_kernel(const float* __restrict__ x,
                                                   const float* __restrict__ wg,
                                                   int* __restrict__ counts,
                                                   int* __restrict__ tok_list,
                                                   float* __restrict__ gate_list) {
  const int wave = threadIdx.x >> 5;
  const int lane = threadIdx.x & 31;
  const int b = blockIdx.x * 8 + wave;
  const float* xr = x + (size_t)b * DIM;

  float acc[NE];
#pragma unroll
  for (int e = 0; e < NE; ++e) acc[e] = 0.0f;

  for (int d = lane; d < DIM; d += 32) {
    const float xv = xr[d];
    const float4 wa = ((const float4*)(wg + d * NE))[0];
    const float4 wb = ((const float4*)(wg + d * NE))[1];
    acc[0] = fmaf(xv, wa.x, acc[0]);
    acc[1] = fmaf(xv, wa.y, acc[1]);
    acc[2] = fmaf(xv, wa.z, acc[2]);
    acc[3] = fmaf(xv, wa.w, acc[3]);
    acc[4] = fmaf(xv, wb.x, acc[4]);
    acc[5] = fmaf(xv, wb.y, acc[5]);
    acc[6] = fmaf(xv, wb.z, acc[6]);
    acc[7] = fmaf(xv, wb.w, acc[7]);
  }
#pragma unroll
  for (int e = 0; e < NE; ++e) {
    float v = acc[e];
#pragma unroll
    for (int off = 16; off > 0; off >>= 1) v += __shfl_xor(v, off, 32);
    acc[e] = v;
  }
  if (lane == 0) {
    int i1 = 0; float v1 = acc[0];
#pragma unroll
    for (int e = 1; e < NE; ++e) if (acc[e] > v1) { v1 = acc[e]; i1 = e; }
    int i2 = -1; float v2 = -INFINITY;
#pragma unroll
    for (int e = 0; e < NE; ++e) if (e != i1 && acc[e] > v2) { v2 = acc[e]; i2 = e; }
    const float t = expf(v2 - v1);
    const float inv = 1.0f / (1.0f + t);
    const float g1 = inv;
    const float g2 = t * inv;
    int p1 = atomicAdd(&counts[i1], 1);
    tok_list[i1 * B_TOK + p1] = b;
    gate_list[i1 * B_TOK + p1] = g1;
    int p2 = atomicAdd(&counts[i2], 1);
    tok_list[i2 * B_TOK + p2] = b;
    gate_list[i2 * B_TOK + p2] = g2;
  }
}

// ---------------------------------------------------------------------------
// Kernel 2: per-expert 3-GEMM chain on a 32-token tile (f32 WMMA).
//   H = relu(X @ W1^T + b1); O = H @ W2^T + b2; M = O @ Wm^T; y += g*M
// X/W1 chunks: double-buffered async global->LDS. W2/Wm: TDM tensor loads
// issued at kernel start, waited right before their consuming phase.
// ---------------------------------------------------------------------------
#define XS_S 36    // 144 B rows (16B aligned for b128 async stores)
#define W1_S 36
#define HS_S 258   // 1032 B rows (8B aligned for v2f reads)
#define W2_S 258   // TDM pads 2 DWORDs per 256 (interval=7, amount=1)
#define OS_S 66
#define WM_S 66    // TDM pads 2 DWORDs per 64  (interval=5, amount=1)

__global__ __launch_bounds__(256) void moe_kernel(
    const float* __restrict__ x,
    const float* __restrict__ W1, const float* __restrict__ b1,
    const float* __restrict__ W2, const float* __restrict__ b2,
    const float* __restrict__ Wm,
    const int* __restrict__ counts,
    const int* __restrict__ tok_list,
    const float* __restrict__ gate_list,
    float* __restrict__ y) {
  __shared__ __align__(16) float Xs[2][TM][XS_S];
  __shared__ __align__(16) float W1s[2][HID][W1_S];
  __shared__ __align__(16) float Hs[TM][HS_S];
  __shared__ __align__(16) float W2s[NC][W2_S];
  __shared__ __align__(16) float Os[TM][OS_S];
  __shared__ __align__(16) float Wms[NT][WM_S];
  __shared__ int   toks[TM];
  __shared__ float gts[TM];

  const int e = blockIdx.y;
  const int count = counts[e];
  const int row0 = blockIdx.x * TM;
  if (row0 >= count) return;   // uniform per-block exit

  const int tid = threadIdx.x;
  const int wv = tid >> 5;
  const int lane = tid & 31;
  const int lrow = lane & 15;          // M (A/C) or N (B/C) index within tile
  const int khalf = (lane >> 4) * 2;   // lanes 0-15 -> K0,K1; 16-31 -> K2,K3
  const int rg = wv & 1;               // 16-token row group
  const int cq = wv >> 1;              // column quadrant 0..3

  const float* W1e = W1 + (size_t)e * HID * DIM;
  const float* W2e = W2 + (size_t)e * NC * HID;
  const float* Wme = Wm + (size_t)e * NT * NC;

  if (tid < TM) {
    const int r = row0 + tid;
    const int rc = (r < count) ? r : (count - 1);  // padded rows -> gate 0
    toks[tid] = tok_list[e * B_TOK + rc];
    gts[tid]  = (r < count) ? gate_list[e * B_TOK + rc] : 0.0f;
  }
  __syncthreads();

  // Kick off the W2 / Wm tile DMAs (TDM); they complete during GEMM1.
  if (wv == 0) {
    tdm_load_2d(W2e, (unsigned)(uintptr_t)&W2s[0][0], HID, NC, HID, 7u, 1u);
    tdm_load_2d(Wme, (unsigned)(uintptr_t)&Wms[0][0], NC, NT, NC, 5u, 1u);
  }

  const v8f zero8 = {0.f, 0.f, 0.f, 0.f, 0.f, 0.f, 0.f, 0.f};

  // ---------------- GEMM1: Hs[32x256] = relu(X @ W1e^T + b1) ----------------
  v8f acc1[4];
#pragma unroll
  for (int ct = 0; ct < 4; ++ct) acc1[ct] = zero8;

  const int r  = tid >> 3;
  const int sg = (tid & 7) * 4;
  const float* xrow = x + (size_t)toks[r] * DIM + sg;  // toks valid after barrier

  // issue chunk 0 into buffer 0 (9 async b128 per wave)
  async_copy16(xrow, &Xs[0][r][sg]);
#pragma unroll
  for (int i = 0; i < 8; ++i) {
    const int j = r + i * 32;
    async_copy16(W1e + (size_t)j * DIM + sg, &W1s[0][j][sg]);
  }

  for (int c = 0; c < NCHUNK; ++c) {
    const int p = c & 1;
    if (c + 1 < NCHUNK) {
      const int kb = (c + 1) * 32;
      async_copy16(xrow + kb, &Xs[1 - p][r][sg]);
#pragma unroll
      for (int i = 0; i < 8; ++i) {
        const int j = r + i * 32;
        async_copy16(W1e + (size_t)j * DIM + kb + sg, &W1s[1 - p][j][sg]);
      }
      WAIT_ASYNC(9);   // in-order retire => chunk c fully resident
    } else {
      WAIT_ASYNC(0);
    }
    __syncthreads();
#pragma unroll
    for (int ks = 0; ks < 8; ++ks) {
      const int k = ks * 4 + khalf;
      const v2f a = *(const v2f*)&Xs[p][rg * 16 + lrow][k];
#pragma unroll
      for (int ct = 0; ct < 4; ++ct) {
        const v2f bb = *(const v2f*)&W1s[p][cq * 64 + ct * 16 + lrow][k];
        acc1[ct] = WMMA_F32(a, bb, acc1[ct]);
      }
    }
    __syncthreads();
  }
  {
    const float* b1e = b1 + e * HID;
#pragma unroll
    for (int ct = 0; ct < 4; ++ct) {
      const int col = cq * 64 + ct * 16 + lrow;
      const float bias = b1e[col];
#pragma unroll
      for (int i = 0; i < 8; ++i) {
        const int row = rg * 16 + (lane >> 4) * 8 + i;  // C layout: M = i / i+8
        const float v = acc1[ct][i] + bias;
        Hs[row][col] = v > 0.0f ? v : 0.0f;
      }
    }
  }
  WAIT_TENSOR(1);   // oldest TDM (W2 tile) complete
  __syncthreads();

  // ---------------- GEMM2: Os[32x64] = Hs @ W2e^T + b2 ----------------
  v8f acc2 = zero8;
#pragma unroll 8
  for (int ks = 0; ks < 64; ++ks) {
    const int k = ks * 4 + khalf;
    const v2f a  = *(const v2f*)&Hs[rg * 16 + lrow][k];
    const v2f bb = *(const v2f*)&W2s[cq * 16 + lrow][k];
    acc2 = WMMA_F32(a, bb, acc2);
  }
  {
    const float* b2e = b2 + e * NC;
    const int col = cq * 16 + lrow;
    const float bias = b2e[col];
#pragma unroll
    for (int i = 0; i < 8; ++i) {
      const int row = rg * 16 + (lane >> 4) * 8 + i;
      Os[row][col] = acc2[i] + bias;
    }
  }
  WAIT_TENSOR(0);   // Wm tile complete
  __syncthreads();

  // ---------------- GEMM3: M[32x512] = Os @ Wme^T ; y += gate * M ----------
  for (int j = 0; j < 8; ++j) {
    const int cb3 = cq * 128 + j * 16;
    v8f acc3 = zero8;
#pragma unroll
    for (int ks = 0; ks < 16; ++ks) {
      const int k = ks * 4 + khalf;
      const v2f a  = *(const v2f*)&Os[rg * 16 + lrow][k];
      const v2f bb = *(const v2f*)&Wms[cb3 + lrow][k];
      acc3 = WMMA_F32(a, bb, acc3);
    }
    const int col = cb3 + lrow;
#pragma unroll
    for (int i = 0; i < 8; ++i) {
      const int row = rg * 16 + (lane >> 4) * 8 + i;
      const float g = gts[row];
      if (g != 0.0f) {
        atomicAdd(&y[(size_t)toks[row] * NT + col], g * acc3[i]);
      }
    }
  }
}

// ---------------------------------------------------------------------------
// Kernel 3: combined[combined == 0] = float64 eps (as float32)
// ---------------------------------------------------------------------------
__global__ __launch_bounds__(256) void finalize_kernel(float* __restrict__ y, int n) {
  int i = blockIdx.x * 256 + threadIdx.x;
  if (i < n) {
    const float v = y[i];
    y[i] = (v == 0.0f) ? 2.2204460492503131e-16f : v;
  }
}

// ---------------------------------------------------------------------------
extern "C" void kernel_launch(void* const* d_in, const int* in_sizes, int n_in,
                              void* d_out, int out_size, void* d_ws, size_t ws_size,
                              hipStream_t stream) {
  (void)in_sizes; (void)n_in; (void)out_size; (void)ws_size;
  const float* x  = (const float*)d_in[0];
  // d_in[1] = labels (unused by reference forward)
  const float* wg = (const float*)d_in[2];
  const float* W1 = (const float*)d_in[3];
  const float* b1 = (const float*)d_in[4];
  const float* W2 = (const float*)d_in[5];
  const float* b2 = (const float*)d_in[6];
  const float* Wm = (const float*)d_in[7];
  float* y = (float*)d_out;

  char* ws = (char*)d_ws;
  int*   counts    = (int*)ws;                                        // NE ints
  int*   tok_list  = (int*)(ws + 256);                                // NE*B ints
  float* gate_list = (float*)(ws + 256 + (size_t)NE * B_TOK * 4);     // NE*B floats

  const int n = B_TOK * NT;
  zero_kernel<<<(n + 255) / 256, 256, 0, stream>>>(y, counts, n);
  gate_kernel<<<B_TOK / 8, 256, 0, stream>>>(x, wg, counts, tok_list, gate_list);
  moe_kernel<<<dim3(B_TOK / TM, NE), 256, 0, stream>>>(x, W1, b1, W2, b2, Wm,
                                                       counts, tok_list, gate_list, y);
  finalize_kernel<<<(n + 255) / 256, 256, 0, stream>>>(y, n);
}